// Models_21534966022474
// MI455X (gfx1250) — compile-verified
//
#include <hip/hip_runtime.h>

#define NUSERS 100000
#define MITEMS 50000
#define NNODES (NUSERS + MITEMS)
#define EDIM   64

typedef __attribute__((ext_vector_type(2))) float v2f;
typedef __attribute__((ext_vector_type(8))) float v8f;

// ---- init: cur = acc = concat(embed_user_0, embed_item_0); nxt0 = 0 ------
__global__ __launch_bounds__(256) void lgcn_init(const float* __restrict__ eu,
                                                 const float* __restrict__ ei,
                                                 float* __restrict__ cur,
                                                 float* __restrict__ acc,
                                                 float* __restrict__ nxt0,
                                                 long long n) {
  long long t = blockIdx.x * (long long)blockDim.x + threadIdx.x;
  if (t >= n) return;
  const long long usz = (long long)NUSERS * EDIM;
  float v = (t < usz) ? eu[t] : ei[t - usz];
  cur[t] = v;
  acc[t] = v;
  nxt0[t] = 0.0f;
}

// ---- COO SpMM: one wave32 per edge, lane handles 2 of 64 dims ------------
// Edge index forced wave-uniform -> rows/cols/vals become scalar (s_load),
// leaving 1 vector gather + 2 f32 atomics per lane. All traffic is
// L2-resident (38.4MB table << 192MB L2).
__global__ __launch_bounds__(256) void lgcn_spmm(const float* __restrict__ src,
                                                 float* __restrict__ dst,
                                                 const int* __restrict__ rows,
                                                 const int* __restrict__ cols,
                                                 const float* __restrict__ vals,
                                                 int E) {
  long long t = blockIdx.x * (long long)blockDim.x + threadIdx.x;
  int e = __builtin_amdgcn_readfirstlane((int)(t >> 5));   // wave-uniform -> SGPR
  if (e >= E) return;                                      // uniform branch
  int lane = threadIdx.x & 31;
  int r = rows[e];                                         // s_load
  int c = cols[e];                                         // s_load
  float v = vals[e];                                       // s_load
  const float2 s = *(const float2*)(src + (size_t)c * EDIM + lane * 2);
  float* d = dst + (size_t)r * EDIM + lane * 2;
  unsafeAtomicAdd(d + 0, v * s.x);   // global_atomic_add_f32, no return
  unsafeAtomicAdd(d + 1, v * s.y);
}

// ---- acc += layer; zero the buffer that becomes next layer's dest --------
__global__ __launch_bounds__(256) void lgcn_addacc_zero(float* __restrict__ acc,
                                                        const float* __restrict__ layer,
                                                        float* __restrict__ oldcur,
                                                        long long n) {
  long long t = blockIdx.x * (long long)blockDim.x + threadIdx.x;
  if (t < n) {
    acc[t] += layer[t];
    oldcur[t] = 0.0f;
  }
}

// ---- batched dot products via V_WMMA_F32_16X16X4_F32 ---------------------
// One wave32 per tile of 16 batch rows. A = u_e (16xK slice), B = i_e^T.
// Diagonal of the 16x16 f32 accumulator = the 16 dot products.
__global__ __launch_bounds__(256) void lgcn_pred(const float* __restrict__ acc,
                                                 const int* __restrict__ user,
                                                 const int* __restrict__ item_i,
                                                 const int* __restrict__ item_j,
                                                 float* __restrict__ out_i,
                                                 float* __restrict__ out_j,
                                                 int batch) {
  const int lane = threadIdx.x & 31;
  const int wave = threadIdx.x >> 5;
  const int tile = blockIdx.x * (blockDim.x >> 5) + wave;
  const int ntiles = (batch + 15) >> 4;
  if (tile >= ntiles) return;            // whole-wave uniform exit; EXEC all-1 inside

  const int m     = lane & 15;           // M for A-lanes, N for B-lanes
  const int khalf = (lane >> 4) << 1;    // K sub-offset: 0 (lanes 0-15) / 2 (lanes 16-31)
  int b = tile * 16 + m;
  if (b >= batch) b = batch - 1;         // clamp padding lanes (stores are guarded)

  const float* urow = acc + (size_t)user[b] * EDIM;
  const float* irow = acc + ((size_t)NUSERS + (size_t)item_i[b]) * EDIM;
  const float* jrow = acc + ((size_t)NUSERS + (size_t)item_j[b]) * EDIM;

  const float scale = 0.25f;             // light_out = acc / (N_LAYERS+1); 0.25*0.25 per product
  v8f ci = {};
  v8f cj = {};
#pragma unroll
  for (int kc = 0; kc < EDIM; kc += 4) {
    v2f a, bi, bj;
    // A 16x4 f32 layout: lane m, VGPR v -> A[m][khalf+v]
    a[0]  = urow[kc + khalf + 0] * scale;
    a[1]  = urow[kc + khalf + 1] * scale;
    // B 4x16 f32 layout: lane n, VGPR v -> B[khalf+v][n] = i_e[n][khalf+v]
    bi[0] = irow[kc + khalf + 0] * scale;
    bi[1] = irow[kc + khalf + 1] * scale;
    bj[0] = jrow[kc + khalf + 0] * scale;
    bj[1] = jrow[kc + khalf + 1] * scale;
    ci = __builtin_amdgcn_wmma_f32_16x16x4_f32(false, a, false, bi, (short)0, ci, false, false);
    cj = __builtin_amdgcn_wmma_f32_16x16x4_f32(false, a, false, bj, (short)0, cj, false, false);
  }

  // C/D layout: lanes 0-15 hold D[v][lane] (v=0..7); lanes 16-31 hold D[v+8][lane-16].
  // Diagonal D[m][m]: lanes 0-7 -> c[lane]; lanes 24-31 -> c[lane-24].
  bool have;
  int vsel, n;
  if (lane < 8)        { have = true;  vsel = lane;      n = lane;      }
  else if (lane >= 24) { have = true;  vsel = lane - 24; n = lane - 16; }
  else                 { have = false; vsel = 0;         n = 0;         }
  int ob = tile * 16 + n;
  if (have && ob < batch) {
    out_i[ob] = ci[vsel];
    out_j[ob] = cj[vsel];
  }
}

// ---- reg loss: 0.5 * (|u0|^2 + |i0|^2 + |j0|^2) / batch -------------------
__global__ __launch_bounds__(256) void lgcn_reg(const float* __restrict__ eu0,
                                                const float* __restrict__ ei0,
                                                const int* __restrict__ user,
                                                const int* __restrict__ item_i,
                                                const int* __restrict__ item_j,
                                                float* __restrict__ out,
                                                int batch) {
  __shared__ float red[8];
  int b = blockIdx.x * blockDim.x + threadIdx.x;
  float s = 0.0f;
  if (b < batch) {
    const float* u = eu0 + (size_t)user[b] * EDIM;
    const float* i = ei0 + (size_t)item_i[b] * EDIM;
    const float* j = ei0 + (size_t)item_j[b] * EDIM;
    for (int d = 0; d < EDIM; d += 2) {
      float2 uv = *(const float2*)(u + d);
      float2 iv = *(const float2*)(i + d);
      float2 jv = *(const float2*)(j + d);
      s += uv.x * uv.x + uv.y * uv.y + iv.x * iv.x + iv.y * iv.y + jv.x * jv.x + jv.y * jv.y;
    }
  }
  for (int off = 16; off > 0; off >>= 1) s += __shfl_down(s, off, 32);
  int lane = threadIdx.x & 31, wave = threadIdx.x >> 5;
  if (lane == 0) red[wave] = s;
  __syncthreads();
  if (threadIdx.x == 0) {
    float tot = 0.0f;
    int nw = blockDim.x >> 5;
    for (int w = 0; w < nw; ++w) tot += red[w];
    unsafeAtomicAdd(out, tot * (0.5f / (float)batch));
  }
}

__global__ void lgcn_zero1(float* p) {
  if (threadIdx.x == 0 && blockIdx.x == 0) *p = 0.0f;
}

extern "C" void kernel_launch(void* const* d_in, const int* in_sizes, int n_in,
                              void* d_out, int out_size, void* d_ws, size_t ws_size,
                              hipStream_t stream) {
  (void)n_in; (void)out_size; (void)ws_size;
  const float* eu0    = (const float*)d_in[0];
  const float* ei0    = (const float*)d_in[1];
  const int*   rows   = (const int*)d_in[2];
  const int*   cols   = (const int*)d_in[3];
  const float* vals   = (const float*)d_in[4];
  const int*   user   = (const int*)d_in[5];
  const int*   item_i = (const int*)d_in[6];
  const int*   item_j = (const int*)d_in[7];
  const int E     = in_sizes[2];
  const int batch = in_sizes[5];

  const long long nelem = (long long)NNODES * EDIM;   // 9.6M floats / buffer
  float* bufA = (float*)d_ws;
  float* bufB = bufA + nelem;
  float* accb = bufB + nelem;

  const int TB = 256;
  dim3 blk(TB);
  unsigned gElem = (unsigned)((nelem + TB - 1) / TB);

  // cur = acc = embeddings; first destination buffer (bufB) zeroed here
  lgcn_init<<<dim3(gElem), blk, 0, stream>>>(eu0, ei0, bufA, accb, bufB, nelem);

  float* cur = bufA;
  float* nxt = bufB;
  long long spmmThreads = (long long)E * 32;
  unsigned gSpmm = (unsigned)((spmmThreads + TB - 1) / TB);
  for (int layer = 0; layer < 3; ++layer) {
    lgcn_spmm<<<dim3(gSpmm), blk, 0, stream>>>(cur, nxt, rows, cols, vals, E);
    // acc += nxt; zero cur (it becomes the destination of the next layer)
    lgcn_addacc_zero<<<dim3(gElem), blk, 0, stream>>>(accb, nxt, cur, nelem);
    float* tmp = cur; cur = nxt; nxt = tmp;
  }

  float* out_i = (float*)d_out;
  float* out_j = out_i + batch;
  float* out_r = out_i + 2 * batch;

  int ntiles = (batch + 15) / 16;
  int wpb = TB / 32;
  unsigned gPred = (unsigned)((ntiles + wpb - 1) / wpb);
  lgcn_pred<<<dim3(gPred), blk, 0, stream>>>(accb, user, item_i, item_j, out_i, out_j, batch);

  lgcn_zero1<<<dim3(1), dim3(32), 0, stream>>>(out_r);
  unsigned gReg = (unsigned)((batch + TB - 1) / TB);
  lgcn_reg<<<dim3(gReg), blk, 0, stream>>>(eu0, ei0, user, item_i, item_j, out_r, batch);
}